// NeuralQKM_69578470195742
// MI455X (gfx1250) — compile-verified
//
#include <hip/hip_runtime.h>

#define NQ    12
#define DIM   4096
#define NL    5
#define BATCH 4096
#define BK    32
#define TS    64
#define LSTR  36                 // padded LDS row stride (floats)
#define TILEF (TS * LSTR)        // floats per 64x32 tile (padded)
#define BUFF  (4 * TILEF)        // floats per buffer (aR,aI,bR,bI)

typedef __attribute__((ext_vector_type(2))) float v2f;
typedef __attribute__((ext_vector_type(8))) float v8f;

// ---------------------------------------------------------------------------
// Kernel 1: QNN state evolution. One block (256 threads) per batch element.
// ---------------------------------------------------------------------------
__global__ __launch_bounds__(256) void qnn_states_kernel(
    const float* __restrict__ X,      // [B][NQ]
    const float* __restrict__ P,      // [NL][NQ][3]
    float* __restrict__ Sr,           // [B][DIM]
    float* __restrict__ Si)           // [B][DIM]
{
  __shared__ float sRe[DIM];
  __shared__ float sIm[DIM];
  const int b = blockIdx.x;
  const int t = threadIdx.x;

  for (int i = t; i < DIM; i += 256) {
    sRe[i] = (i == 0) ? 1.0f : 0.0f;
    sIm[i] = 0.0f;
  }

  for (int l = 0; l < NL; ++l) {
    for (int q = 0; q < NQ; ++q) {
      const float phi = P[(l * NQ + q) * 3 + 0];
      const float th  = P[(l * NQ + q) * 3 + 1];
      const float lam = P[(l * NQ + q) * 3 + 2];
      float sth, cth;  sincosf(0.5f * th, &sth, &cth);
      float sp, cp;    sincosf(0.5f * (lam + phi), &sp, &cp);
      float sm_, cm;   sincosf(0.5f * (lam - phi), &sm_, &cm);

      float u00r =  cth * cp, u00i = -cth * sp;
      float u01r = -sth * cm, u01i =  sth * sm_;
      float u10r =  sth * cm, u10i =  sth * sm_;
      float u11r =  cth * cp, u11i =  cth * sp;

      if (l == NL - 1) {
        float gs, gc; sincosf(0.5f * X[b * NQ + q], &gs, &gc);
        float m00r = gc * u00r - gs * u10r, m00i = gc * u00i - gs * u10i;
        float m01r = gc * u01r - gs * u11r, m01i = gc * u01i - gs * u11i;
        float m10r = gs * u00r + gc * u10r, m10i = gs * u00i + gc * u10i;
        float m11r = gs * u01r + gc * u11r, m11i = gs * u01i + gc * u11i;
        u00r = m00r; u00i = m00i; u01r = m01r; u01i = m01i;
        u10r = m10r; u10i = m10i; u11r = m11r; u11i = m11i;
      }

      const int stride = 1 << (11 - q);
      __syncthreads();
#pragma unroll
      for (int p = 0; p < 8; ++p) {
        const int pid = p * 256 + t;
        const int i0  = ((pid & ~(stride - 1)) << 1) | (pid & (stride - 1));
        const int i1  = i0 + stride;
        const float a0r = sRe[i0], a0i = sIm[i0];
        const float a1r = sRe[i1], a1i = sIm[i1];
        sRe[i0] = u00r * a0r - u00i * a0i + u01r * a1r - u01i * a1i;
        sIm[i0] = u00r * a0i + u00i * a0r + u01r * a1i + u01i * a1r;
        sRe[i1] = u10r * a0r - u10i * a0i + u11r * a1r - u11i * a1i;
        sIm[i1] = u10r * a0i + u10i * a0r + u11r * a1i + u11i * a1r;
      }
    }

    for (int q = 0; q < NQ - 1; ++q) {
      const int tb = 10 - q;
      const int cb = tb + 1;
      const int lowmask = (1 << tb) - 1;
      __syncthreads();
#pragma unroll
      for (int j = 0; j < 4; ++j) {
        const int k  = j * 256 + t;
        const int i0 = ((k & ~lowmask) << 2) | (1 << cb) | (k & lowmask);
        const int i1 = i0 | (1 << tb);
        float r0 = sRe[i0], r1 = sRe[i1];
        float m0 = sIm[i0], m1 = sIm[i1];
        sRe[i0] = r1; sRe[i1] = r0;
        sIm[i0] = m1; sIm[i1] = m0;
      }
    }
  }

  __syncthreads();
  for (int i = t; i < DIM; i += 256) {
    Sr[(size_t)b * DIM + i] = sRe[i];
    Si[(size_t)b * DIM + i] = sIm[i];
  }
}

// ---------------------------------------------------------------------------
// Kernel 2: K = |S S^H|^2 via FP32 WMMA + async global->LDS double buffering.
// Only tiles with bx >= by computed (K symmetric); both halves written.
// ---------------------------------------------------------------------------
__global__ __launch_bounds__(256) void gram_kernel(
    const float* __restrict__ Sr,
    const float* __restrict__ Si,
    float* __restrict__ K)
{
  const int bx = blockIdx.x, by = blockIdx.y;
  if (bx < by) return;   // uniform: EXEC stays all-ones for WMMA

  __shared__ float lds[2 * BUFF];   // [buf][tile: aR,aI,bR,bI][64][LSTR]

  const int t    = threadIdx.x;
  const int wave = t >> 5;
  const int lane = t & 31;
  const int rowBase = by * TS;
  const int colBase = bx * TS;

  const int sn  = wave & 3;
  const int sm0 = wave >> 2;
  const int sm1 = sm0 + 2;

  const int m     = lane & 15;
  const int khalf = (lane >> 4) << 1;

  // Async staging: each thread moves 2 float4 per 64x32 tile (512 total).
  const int r0 = t >> 3,          c0 = (t & 7) << 2;
  const int r1 = (256 + t) >> 3,  c1 = ((256 + t) & 7) << 2;

  const float* gAR0 = &Sr[(size_t)(rowBase + r0) * DIM + c0];
  const float* gAI0 = &Si[(size_t)(rowBase + r0) * DIM + c0];
  const float* gBR0 = &Sr[(size_t)(colBase + r0) * DIM + c0];
  const float* gBI0 = &Si[(size_t)(colBase + r0) * DIM + c0];
  const float* gAR1 = &Sr[(size_t)(rowBase + r1) * DIM + c1];
  const float* gAI1 = &Si[(size_t)(rowBase + r1) * DIM + c1];
  const float* gBR1 = &Sr[(size_t)(colBase + r1) * DIM + c1];
  const float* gBI1 = &Si[(size_t)(colBase + r1) * DIM + c1];

  // LDS byte addresses (flat scheme: low 32 bits of generic pointer = LDS addr)
  const unsigned ldsBase = (unsigned)(uintptr_t)&lds[0];
  const unsigned d0 = ldsBase + (unsigned)(r0 * LSTR + c0) * 4u;
  const unsigned d1 = ldsBase + (unsigned)(r1 * LSTR + c1) * 4u;

#define ASYNC_LD(DST, SRC)                                                   \
  asm volatile("global_load_async_to_lds_b128 %0, %1, off"                   \
               :: "v"(DST), "v"(SRC) : "memory")

#define ISSUE(KB, BUFB)                                                      \
  do {                                                                       \
    ASYNC_LD(d0 + (BUFB) + 0u * (TILEF * 4u), gAR0 + (KB));                  \
    ASYNC_LD(d0 + (BUFB) + 1u * (TILEF * 4u), gAI0 + (KB));                  \
    ASYNC_LD(d0 + (BUFB) + 2u * (TILEF * 4u), gBR0 + (KB));                  \
    ASYNC_LD(d0 + (BUFB) + 3u * (TILEF * 4u), gBI0 + (KB));                  \
    ASYNC_LD(d1 + (BUFB) + 0u * (TILEF * 4u), gAR1 + (KB));                  \
    ASYNC_LD(d1 + (BUFB) + 1u * (TILEF * 4u), gAI1 + (KB));                  \
    ASYNC_LD(d1 + (BUFB) + 2u * (TILEF * 4u), gBR1 + (KB));                  \
    ASYNC_LD(d1 + (BUFB) + 3u * (TILEF * 4u), gBI1 + (KB));                  \
  } while (0)

#define WAIT_ASYNC_8() asm volatile("s_wait_asynccnt 0x8" ::: "memory")
#define WAIT_ASYNC_0() asm volatile("s_wait_asynccnt 0x0" ::: "memory")

  v8f accRe0 = {}, accP0 = {}, accQ0 = {};
  v8f accRe1 = {}, accP1 = {}, accQ1 = {};

#define COMPUTE(BASE)                                                        \
  do {                                                                       \
    const float* aR_ = (BASE);                                               \
    const float* aI_ = (BASE) + TILEF;                                       \
    const float* bR_ = (BASE) + 2 * TILEF;                                   \
    const float* bI_ = (BASE) + 3 * TILEF;                                   \
    _Pragma("unroll")                                                        \
    for (int kk = 0; kk < BK; kk += 4) {                                     \
      const int ko = kk + khalf;                                             \
      const v2f br  = *(const v2f*)&bR_[(sn  * 16 + m) * LSTR + ko];         \
      const v2f bi  = *(const v2f*)&bI_[(sn  * 16 + m) * LSTR + ko];         \
      const v2f ar0 = *(const v2f*)&aR_[(sm0 * 16 + m) * LSTR + ko];         \
      const v2f ai0 = *(const v2f*)&aI_[(sm0 * 16 + m) * LSTR + ko];         \
      const v2f ar1 = *(const v2f*)&aR_[(sm1 * 16 + m) * LSTR + ko];         \
      const v2f ai1 = *(const v2f*)&aI_[(sm1 * 16 + m) * LSTR + ko];         \
      accRe0 = __builtin_amdgcn_wmma_f32_16x16x4_f32(false, ar0, false, br, (short)0, accRe0, false, false); \
      accRe0 = __builtin_amdgcn_wmma_f32_16x16x4_f32(false, ai0, false, bi, (short)0, accRe0, false, false); \
      accP0  = __builtin_amdgcn_wmma_f32_16x16x4_f32(false, ai0, false, br, (short)0, accP0,  false, false); \
      accQ0  = __builtin_amdgcn_wmma_f32_16x16x4_f32(false, ar0, false, bi, (short)0, accQ0,  false, false); \
      accRe1 = __builtin_amdgcn_wmma_f32_16x16x4_f32(false, ar1, false, br, (short)0, accRe1, false, false); \
      accRe1 = __builtin_amdgcn_wmma_f32_16x16x4_f32(false, ai1, false, bi, (short)0, accRe1, false, false); \
      accP1  = __builtin_amdgcn_wmma_f32_16x16x4_f32(false, ai1, false, br, (short)0, accP1,  false, false); \
      accQ1  = __builtin_amdgcn_wmma_f32_16x16x4_f32(false, ar1, false, bi, (short)0, accQ1,  false, false); \
    }                                                                        \
  } while (0)

  // Prologue: chunk 0 -> buffer 0 (8 async loads in flight).
  ISSUE(0, 0u);

  // DIM/BK = 128 chunks (even): process two per iteration (buf0 then buf1).
  for (int kb = 0; kb < DIM; kb += 2 * BK) {
    // Phase A: prefetch chunk kb+BK into buf1, compute chunk kb from buf0.
    ISSUE(kb + BK, (unsigned)(BUFF * 4));
    WAIT_ASYNC_8();                 // chunk kb resident (in-order completion)
    __syncthreads();
    COMPUTE(&lds[0]);
    __syncthreads();                // all waves done with buf0

    // Phase B: prefetch chunk kb+2BK into buf0, compute chunk kb+BK from buf1.
    if (kb + 2 * BK < DIM) { ISSUE(kb + 2 * BK, 0u); WAIT_ASYNC_8(); }
    else                   { WAIT_ASYNC_0(); }
    __syncthreads();
    COMPUTE(&lds[BUFF]);
    __syncthreads();                // all waves done with buf1
  }

  // Epilogue: K = Re^2 + (P - Q)^2; write both symmetric halves.
  const int mo = 8 * (lane >> 4);
  const int no = lane & 15;
  const int col = colBase + sn * 16 + no;
#pragma unroll
  for (int r = 0; r < 8; ++r) {
    {
      const float re = accRe0[r];
      const float im = accP0[r] - accQ0[r];
      const int row = rowBase + sm0 * 16 + r + mo;
      const float v = re * re + im * im;
      K[(size_t)row * BATCH + col] = v;
      K[(size_t)col * BATCH + row] = v;
    }
    {
      const float re = accRe1[r];
      const float im = accP1[r] - accQ1[r];
      const int row = rowBase + sm1 * 16 + r + mo;
      const float v = re * re + im * im;
      K[(size_t)row * BATCH + col] = v;
      K[(size_t)col * BATCH + row] = v;
    }
  }
#undef ASYNC_LD
#undef ISSUE
#undef WAIT_ASYNC_8
#undef WAIT_ASYNC_0
#undef COMPUTE
}

// ---------------------------------------------------------------------------
extern "C" void kernel_launch(void* const* d_in, const int* in_sizes, int n_in,
                              void* d_out, int out_size, void* d_ws, size_t ws_size,
                              hipStream_t stream) {
  const float* X = (const float*)d_in[0];   // [B][NQ] float32
  const float* P = (const float*)d_in[1];   // [NL][NQ][3] float32
  float* Sr = (float*)d_ws;                 // B*DIM floats
  float* Si = Sr + (size_t)BATCH * DIM;     // B*DIM floats
  float* K  = (float*)d_out;                // B*B floats

  qnn_states_kernel<<<BATCH, 256, 0, stream>>>(X, P, Sr, Si);
  gram_kernel<<<dim3(BATCH / TS, BATCH / TS), 256, 0, stream>>>(Sr, Si, K);
}